// Cluster_47614007444072
// MI455X (gfx1250) — compile-verified
//
#include <hip/hip_runtime.h>

typedef __attribute__((ext_vector_type(2))) float v2f;
typedef __attribute__((ext_vector_type(8))) float v8f;

#define HIDDEN   512
#define NB       1024
#define NCLASS   50
#define NCENTER  8
#define MROWS    (NB * NCENTER)   // 8192 flattened (b,c) rows

// ---- CDNA5 async global->LDS staging (ASYNCcnt path), guarded so the build
// ---- falls back to VGPR staging if this toolchain lacks the builtin.
#if defined(__AMDGCN__) && __has_builtin(__builtin_amdgcn_global_load_async_to_lds_b32)
#define USE_ASYNC_LDS 1
#else
#define USE_ASYNC_LDS 0
#endif

#if USE_ASYNC_LDS
// Probe-derived signature: generic int* parameters (per hipcc diagnostic).
#define ASYNC_LDS_B32(gsrc, ldst) \
    __builtin_amdgcn_global_load_async_to_lds_b32((int*)(gsrc), (int*)(ldst), 0, 0)
#if __has_builtin(__builtin_amdgcn_s_wait_asynccnt)
#define ASYNC_WAIT() __builtin_amdgcn_s_wait_asynccnt(0)
#else
#define ASYNC_WAIT() asm volatile("s_wait_asynccnt 0" ::: "memory")
#endif
#endif

// ---------------------------------------------------------------------------
// Kernel 1: mul[r][d] = sum_h (x[r/8][h]*mask[r%8][h]) * W[h][d]
// WMMA f32 16x16x4.  M=8192 N=512 K=512.
// Block 256 thr (8 waves), tile 128x64, K-panel 32.
// Wave w owns a 16x64 strip (4 accumulators); A staged fused with mask.
// LDS panels stored K-pair interleaved so every fragment is one ds_load_b64.
// ---------------------------------------------------------------------------
__global__ __launch_bounds__(256)
void gemm_mul_kernel(const float* __restrict__ x, const float* __restrict__ mask,
                     const float* __restrict__ W, float* __restrict__ mul)
{
    __shared__ v2f lAp[128][17];   // [m][k/2]  (odd pair pitch -> bank stagger)
    __shared__ v2f lBp[16][64];    // [k/2][n]

    const int t    = threadIdx.x;
    const int w    = t >> 5;
    const int lane = t & 31;
    const int lr   = lane & 15;
    const int kh   = lane >> 4;

    const int n0 = blockIdx.x * 64;
    const int m0 = blockIdx.y * 128;
    const int mo = 16 * w;

    v8f acc[4] = {{}, {}, {}, {}};

    for (int k0 = 0; k0 < HIDDEN; k0 += 32) {
        // ---- B panel 32x64 -> pair-interleaved LDS (8 elems/thread)
        #pragma unroll
        for (int i = 0; i < 8; ++i) {
            int idx = t + 256 * i;
            int row = idx >> 6;
            int col = idx & 63;
            const float* gsrc = &W[(k0 + row) * HIDDEN + n0 + col];
            float* ldst = ((float*)&lBp[row >> 1][col]) + (row & 1);
#if USE_ASYNC_LDS
            ASYNC_LDS_B32(gsrc, ldst);
#else
            *ldst = *gsrc;
#endif
        }
        // ---- A panel 128x32, fused x*mask (16 elems/thread) — VGPR path
        #pragma unroll
        for (int i = 0; i < 16; ++i) {
            int idx = t + 256 * i;
            int row = idx >> 5;
            int col = idx & 31;
            int gr  = m0 + row;
            float v = x[(gr >> 3) * HIDDEN + k0 + col] *
                      mask[(gr & 7) * HIDDEN + k0 + col];
            ((float*)&lAp[row][col >> 1])[col & 1] = v;
        }
#if USE_ASYNC_LDS
        ASYNC_WAIT();
#endif
        __syncthreads();

        #pragma unroll
        for (int kk = 0; kk < 32; kk += 4) {
            const int kp = (kk >> 1) + kh;       // pair index for this lane
            v2f a = lAp[mo + lr][kp];
            #pragma unroll
            for (int j = 0; j < 4; ++j) {
                v2f b = lBp[kp][16 * j + lr];
                acc[j] = __builtin_amdgcn_wmma_f32_16x16x4_f32(
                    false, a, false, b, (short)0, acc[j], false, false);
            }
        }
        __syncthreads();
    }

    #pragma unroll
    for (int j = 0; j < 4; ++j)
        #pragma unroll
        for (int v = 0; v < 8; ++v) {
            int row = m0 + mo + v + 8 * kh;      // C/D layout: VGPR v -> M=v / v+8
            mul[row * HIDDEN + n0 + 16 * j + lr] = acc[j][v];
        }
}

// ---------------------------------------------------------------------------
// Kernel 2: row squared-norms of mul (8192 rows) and class_mu (400 rows).
// One wave per 512-element row.
// ---------------------------------------------------------------------------
__global__ __launch_bounds__(256)
void norms_kernel(const float* __restrict__ mul, const float* __restrict__ class_mu,
                  float* __restrict__ nm, float* __restrict__ nmu)
{
    const int rid  = blockIdx.x * 8 + (threadIdx.x >> 5);
    const int lane = threadIdx.x & 31;

    const float* src;
    float* dst;
    if (rid < MROWS) {
        src = mul + (size_t)rid * HIDDEN;
        dst = nm + rid;
    } else {
        int r = rid - MROWS;
        if (r >= NCLASS * NCENTER) return;
        src = class_mu + (size_t)r * HIDDEN;
        dst = nmu + r;
    }

    float s = 0.0f;
    for (int i = lane; i < HIDDEN; i += 32) {
        float v = src[i];
        s += v * v;
    }
    #pragma unroll
    for (int off = 16; off > 0; off >>= 1)
        s += __shfl_xor(s, off, 32);
    if (lane == 0) *dst = s;
}

// ---------------------------------------------------------------------------
// Kernel 3: per-center dot via WMMA, fused Student-t epilogue.
// center c: dot[b][k] = mul_row(b*8+c) . mu_row(k*8+c), M=1024 N=64(pad) K=512
// score = 1/(1 + nm + nmu - 2*dot) -> ws_score[b][k][c]
// ---------------------------------------------------------------------------
__global__ __launch_bounds__(256)
void score_kernel(const float* __restrict__ mul, const float* __restrict__ class_mu,
                  const float* __restrict__ nm, const float* __restrict__ nmu,
                  float* __restrict__ score)
{
    __shared__ v2f lAp[128][17];   // [b-row][k/2]
    __shared__ v2f lBp[16][64];    // [k/2][class]

    const int t    = threadIdx.x;
    const int w    = t >> 5;
    const int lane = t & 31;
    const int lr   = lane & 15;
    const int kh   = lane >> 4;

    const int b0 = blockIdx.x * 128;
    const int c  = blockIdx.y;
    const int mo = 16 * w;

    // zero the zero-padded class columns (50..63) once; never rewritten
    for (int idx = t; idx < 16 * (64 - NCLASS); idx += 256) {
        int row = idx / (64 - NCLASS);
        int col = NCLASS + idx % (64 - NCLASS);
        lBp[row][col] = (v2f){0.0f, 0.0f};
    }

    v8f acc[4] = {{}, {}, {}, {}};

    for (int k0 = 0; k0 < HIDDEN; k0 += 32) {
        // ---- A panel: 128x32 rows of mul for center c (16/thread)
        #pragma unroll
        for (int i = 0; i < 16; ++i) {
            int idx = t + 256 * i;
            int row = idx >> 5;
            int col = idx & 31;
            const float* gsrc =
                &mul[((size_t)(b0 + row) * NCENTER + c) * HIDDEN + k0 + col];
            float* ldst = ((float*)&lAp[row][col >> 1]) + (col & 1);
#if USE_ASYNC_LDS
            ASYNC_LDS_B32(gsrc, ldst);
#else
            *ldst = *gsrc;
#endif
        }
        // ---- B panel: B[k][kc] = class_mu[kc][c][k0+k], kc<50 only (8/thread)
        #pragma unroll
        for (int i = 0; i < 8; ++i) {
            int idx = t + 256 * i;
            int row = idx >> 6;
            int kc  = idx & 63;
            if (kc < NCLASS) {
                const float* gsrc =
                    &class_mu[((size_t)kc * NCENTER + c) * HIDDEN + k0 + row];
                float* ldst = ((float*)&lBp[row >> 1][kc]) + (row & 1);
#if USE_ASYNC_LDS
                ASYNC_LDS_B32(gsrc, ldst);
#else
                *ldst = *gsrc;
#endif
            }
        }
#if USE_ASYNC_LDS
        ASYNC_WAIT();
#endif
        __syncthreads();

        #pragma unroll
        for (int kk = 0; kk < 32; kk += 4) {
            const int kp = (kk >> 1) + kh;
            v2f a = lAp[mo + lr][kp];
            #pragma unroll
            for (int j = 0; j < 4; ++j) {
                v2f b = lBp[kp][16 * j + lr];
                acc[j] = __builtin_amdgcn_wmma_f32_16x16x4_f32(
                    false, a, false, b, (short)0, acc[j], false, false);
            }
        }
        __syncthreads();
    }

    #pragma unroll
    for (int j = 0; j < 4; ++j)
        #pragma unroll
        for (int v = 0; v < 8; ++v) {
            int brow = b0 + mo + v + 8 * kh;
            int kc   = 16 * j + lr;
            if (kc < NCLASS) {
                float d2 = nm[brow * NCENTER + c] + nmu[kc * NCENTER + c]
                           - 2.0f * acc[j][v];
                score[((size_t)brow * NCLASS + kc) * NCENTER + c] =
                    1.0f / (1.0f + d2);
            }
        }
}

// ---------------------------------------------------------------------------
// Kernel 4: normalize over centers. distances = s/(S+eps); k_assign = s/S.
// ---------------------------------------------------------------------------
__global__ __launch_bounds__(256)
void finalize_kernel(const float* __restrict__ score,
                     float* __restrict__ k_assign, float* __restrict__ distances)
{
    int tid = blockIdx.x * blockDim.x + threadIdx.x;
    if (tid >= NB * NCLASS) return;
    const float* s = score + (size_t)tid * NCENTER;
    float v[NCENTER];
    float S = 0.0f;
    #pragma unroll
    for (int c = 0; c < NCENTER; ++c) { v[c] = s[c]; S += v[c]; }
    float inv1 = 1.0f / (S + 1e-8f);
    float inv2 = 1.0f / S;
    #pragma unroll
    for (int c = 0; c < NCENTER; ++c) {
        distances[(size_t)tid * NCENTER + c] = v[c] * inv1;
        k_assign[(size_t)tid * NCENTER + c]  = v[c] * inv2;
    }
}

// ---------------------------------------------------------------------------
extern "C" void kernel_launch(void* const* d_in, const int* in_sizes, int n_in,
                              void* d_out, int out_size, void* d_ws, size_t ws_size,
                              hipStream_t stream)
{
    const float* x        = (const float*)d_in[0];   // [1024,512]
    const float* class_mu = (const float*)d_in[1];   // [50,8,512]
    const float* mask     = (const float*)d_in[2];   // [8,512]
    const float* W        = (const float*)d_in[3];   // [512,512]

    float* out       = (float*)d_out;
    float* mul       = out;                                     // 4194304
    float* k_assign  = out + (size_t)NB * NCENTER * HIDDEN;     // 409600
    float* distances = k_assign + (size_t)NB * NCLASS * NCENTER;// 409600

    float* ws   = (float*)d_ws;
    float* wscr = ws;                                    // 409600 floats
    float* nm   = wscr + (size_t)NB * NCLASS * NCENTER;  // 8192 floats
    float* nmu  = nm + MROWS;                            // 400 floats

    dim3 blk(256);

    dim3 g1(HIDDEN / 64, MROWS / 128);                   // (8, 64)
    gemm_mul_kernel<<<g1, blk, 0, stream>>>(x, mask, W, mul);

    int rows = MROWS + NCLASS * NCENTER;                 // 8592
    norms_kernel<<<dim3((rows + 7) / 8), blk, 0, stream>>>(mul, class_mu, nm, nmu);

    dim3 g3(NB / 128, NCENTER);                          // (8, 8)
    score_kernel<<<g3, blk, 0, stream>>>(mul, class_mu, nm, nmu, wscr);

    int pairs = NB * NCLASS;                             // 51200
    finalize_kernel<<<dim3((pairs + 255) / 256), blk, 0, stream>>>(wscr, k_assign, distances);
}